// FilterAugment_19121194401928
// MI455X (gfx1250) — compile-verified
//
#include <hip/hip_runtime.h>

// FilterAugment: out[b,0,f,t] = features[b,f,t] * 10^(filt_db[b,f]/20)
// B=64, F=256, T=4096, N_BAND=4 (band_bndry: 5 entries, band_factors: Bx5)
//
// Pure HBM-streaming kernel (512 MiB moved, ~1 FLOP per 8 bytes).
// CDNA5 strategy: wave32 blocks, 128-bit global loads/stores with
// non-temporal cache hints (footprint >> 192MB L2, zero reuse),
// block-uniform scalar computation of the per-(b,f) gain.

#define B_DIM 64
#define F_DIM 256
#define T_DIM 4096
#define NBAND 4

typedef __attribute__((ext_vector_type(4))) float v4f;

__global__ __launch_bounds__(256) void filter_augment_kernel(
    const float* __restrict__ features,      // [B, F, T]
    const float* __restrict__ band_factors,  // [B, NBAND+1]
    const int*   __restrict__ band_bndry,    // [NBAND+1]
    float*       __restrict__ out)           // [B, 1, F, T] == [B, F, T]
{
    const int bf = blockIdx.x;        // 0 .. B*F-1 (one block per (b,f) row)
    const int b  = bf >> 8;           // / F_DIM
    const int f  = bf & (F_DIM - 1);  // % F_DIM

    // ---- per-(b,f) gain: uniform across the block -> scalar path ----
    int bndry[NBAND + 1];
#pragma unroll
    for (int i = 0; i <= NBAND; ++i) bndry[i] = band_bndry[i];

    // idx = clip(searchsorted_right(bndry, f) - 1, 0, NBAND-1)
    // bndry[0]==0 and bndry[NBAND]==F, so counting interior crossings suffices.
    int idx = 0;
#pragma unroll
    for (int i = 1; i < NBAND; ++i) idx += (f >= bndry[i]) ? 1 : 0;

    const int   lo    = bndry[idx];
    const int   width = bndry[idx + 1] - lo;
    const float t     = (width > 1)
                          ? (float)(f - lo) / (float)(width - 1)
                          : 0.0f;
    const float s_db  = band_factors[b * (NBAND + 1) + idx];
    const float e_db  = band_factors[b * (NBAND + 1) + idx + 1];
    const float db    = s_db + (e_db - s_db) * t;
    // 10^(db/20) == exp2(db * log2(10)/20)
    const float gain  = exp2f(db * 0.166096404744368117f);

    // ---- stream one row of T=4096 floats: 256 threads x 4 x float4 ----
    const size_t row = (size_t)bf * T_DIM;
    const v4f* __restrict__ src = reinterpret_cast<const v4f*>(features + row);
    v4f*       __restrict__ dst = reinterpret_cast<v4f*>(out + row);

#pragma unroll
    for (int j = 0; j < 4; ++j) {
        const int v = (int)threadIdx.x + j * 256;   // coalesced b128 accesses
        v4f x = __builtin_nontemporal_load(src + v); // global_load_b128 th:NT
        x.x *= gain;
        x.y *= gain;
        x.z *= gain;
        x.w *= gain;
        __builtin_nontemporal_store(x, dst + v);     // global_store_b128 th:NT
    }
}

extern "C" void kernel_launch(void* const* d_in, const int* in_sizes, int n_in,
                              void* d_out, int out_size, void* d_ws, size_t ws_size,
                              hipStream_t stream) {
    const float* features     = (const float*)d_in[0]; // B*F*T
    const float* band_factors = (const float*)d_in[1]; // B*(NBAND+1)
    const int*   band_bndry   = (const int*)d_in[2];   // NBAND+1
    float*       out          = (float*)d_out;         // B*1*F*T

    const int nblocks = B_DIM * F_DIM; // 16384 rows, one per block
    filter_augment_kernel<<<nblocks, 256, 0, stream>>>(
        features, band_factors, band_bndry, out);
}